// UnifiedTripleLoss_60318520705735
// MI455X (gfx1250) — compile-verified
//
#include <hip/hip_runtime.h>
#include <math.h>

// ------------------- problem constants -------------------
#define Bn     32
#define Hn     1024
#define Wn     1024
#define HW     (Hn * Wn)          // 1<<20
#define NPIX   (Bn * HW)          // 32M
#define SPTS   1000               // SAMPLE_POINTS
#define SPAD   1008               // padded to 63*16 for 16x16 WMMA tiles
#define NTILE  63
#define SEG    ((HW + SPTS - 1) / SPTS)   // 1049
#define THRESH 0.1f
#define MAXD   0.1f
#define SMOOTH 1e-5f
#define DIAG   1448.1546f         // sqrt(1024^2 + 1024^2)
#define BIGF   3.0e38f

// ------------- workspace layout (4-byte word offsets) -------------
#define F_INTER 0                  // [Bn]  dice intersection
#define F_PSUM  32                 // [Bn]  sum probs
#define F_TSUM  64                 // [Bn]  sum target
#define F_BSUM  96                 // [2]   weighted_err sum, (pb-tb)^2 sum
#define F_AMAX  100                // [Bn*2] directed hausdorff maxima (overwritten, no init)
#define U_PCNT  256                // [Bn]  boundary pixel counts (pred map)
#define U_TCNT  288                // [Bn]  boundary pixel counts (target map)
#define U_PMASK 512                           // [NPIX/32] bitmask pb>thresh
#define U_TMASK (U_PMASK + NPIX / 32)         // [NPIX/32] bitmask tb>thresh
#define F_PPTS  (U_TMASK + NPIX / 32)         // [Bn*SPAD*2] (y,x) floats
#define F_TPTS  (F_PPTS + Bn * SPAD * 2)
#define U_PVAL  (F_TPTS + Bn * SPAD * 2)      // [Bn*SPAD] validity
#define U_TVAL  (U_PVAL + Bn * SPAD)
// total ~ (U_TVAL + Bn*SPAD) * 4 bytes ~= 9.2 MB of d_ws

typedef __attribute__((ext_vector_type(2))) float v2f;
typedef __attribute__((ext_vector_type(8))) float v8f;

__device__ __forceinline__ float sigmoidf_(float v) { return 1.0f / (1.0f + __expf(-v)); }

__device__ __forceinline__ float waveRedAdd(float v) {
  #pragma unroll
  for (int o = 16; o; o >>= 1) v += __shfl_down(v, o, 32);
  return v;
}

// ------------------- kernel 0: zero accumulators -------------------
__global__ void k_init(unsigned* wsU) {
  int t = threadIdx.x;
  if (t < 512) wsU[t] = 0u;   // covers F_INTER..F_AMAX and U_PCNT/U_TCNT
}

// ------------------- kernel 1: fused sigmoid + sobel + sums + masks -------------------
// One block = 256-pixel row strip. Halo tiles (3 rows x 258 cols) for target and
// sigmoid(pred) staged in LDS; stencil reads hit ds_load instead of global.
#define TW 258                      // 256 + 2 halo columns
__global__ void k_fused(const float* __restrict__ pred, const float* __restrict__ targ,
                        float* wsF, unsigned* wsU) {
  __shared__ float sT[3 * TW];
  __shared__ float sP[3 * TW];

  const int tid  = threadIdx.x;
  const int idx0 = blockIdx.x * 256;
  const int b    = idx0 >> 20;
  const int rem  = idx0 & (HW - 1);
  const int y    = rem >> 10;
  const int x0   = rem & (Wn - 1);          // 0,256,512,768
  const float* Pb = pred + ((size_t)b << 20);
  const float* Tb = targ + ((size_t)b << 20);

  // stage halo tile (zero-padded SAME boundary); sigmoid applied at load
  for (int i = tid; i < 3 * TW; i += 256) {
    const int r  = i / TW;                  // 0..2  -> rows y-1..y+1
    const int c  = i - r * TW;              // 0..257 -> cols x0-1..x0+256
    const int yy = y + r - 1;
    const int xx = x0 + c - 1;
    const bool in = ((unsigned)yy < (unsigned)Hn) && ((unsigned)xx < (unsigned)Wn);
    const size_t o = ((size_t)(yy & (Hn - 1)) << 10) + (xx & (Wn - 1));
    sT[i] = in ? Tb[o] : 0.0f;
    sP[i] = in ? sigmoidf_(Pb[o]) : 0.0f;
  }
  // prefetch two rows ahead (global_prefetch_b8 path); cheap latency hiding
  if (y + 2 < Hn) __builtin_prefetch(&Pb[((size_t)(y + 2) << 10) + x0 + tid], 0, 1);
  __syncthreads();

  const int cc = tid + 1;                   // center column in tile
  float tv[9], pv[9];
  #pragma unroll
  for (int dy = 0; dy < 3; ++dy) {
    #pragma unroll
    for (int dx = 0; dx < 3; ++dx) {
      tv[dy * 3 + dx] = sT[dy * TW + cc + dx - 1];
      pv[dy * 3 + dx] = sP[dy * TW + cc + dx - 1];
    }
  }
  // Sobel (cross-correlation, matches lax.conv_general_dilated)
  float gxT = (-tv[0] + tv[2]) + (-2.f * tv[3] + 2.f * tv[5]) + (-tv[6] + tv[8]);
  float gyT = (-tv[0] - 2.f * tv[1] - tv[2]) + (tv[6] + 2.f * tv[7] + tv[8]);
  float gxP = (-pv[0] + pv[2]) + (-2.f * pv[3] + 2.f * pv[5]) + (-pv[6] + pv[8]);
  float gyP = (-pv[0] - 2.f * pv[1] - pv[2]) + (pv[6] + 2.f * pv[7] + pv[8]);
  const float tb = sqrtf(gxT * gxT + gyT * gyT + 1e-8f);
  const float pb = sqrtf(gxP * gxP + gyP * gyP + 1e-8f);

  const float probs = pv[4];
  const float tc    = tv[4];
  const float we    = fabsf(probs - tc) * (tb * 2.0f + 1.0f);
  const float se    = (pb - tb) * (pb - tb);

  // wave32 reductions, one atomic per wave per quantity
  const int lane = tid & 31;
  const int idx  = idx0 + tid;
  float s;
  s = waveRedAdd(probs * tc); if (lane == 0) atomicAdd(&wsF[F_INTER + b], s);
  s = waveRedAdd(probs);      if (lane == 0) atomicAdd(&wsF[F_PSUM  + b], s);
  s = waveRedAdd(tc);         if (lane == 0) atomicAdd(&wsF[F_TSUM  + b], s);
  s = waveRedAdd(we);         if (lane == 0) atomicAdd(&wsF[F_BSUM + 0], s);
  s = waveRedAdd(se);         if (lane == 0) atomicAdd(&wsF[F_BSUM + 1], s);

  // boundary bitmasks via wave32 ballot (word per 32 consecutive x)
  unsigned pm = (unsigned)__ballot(pb > THRESH);
  unsigned tm = (unsigned)__ballot(tb > THRESH);
  if (lane == 0) {
    wsU[U_PMASK + (idx >> 5)] = pm;
    wsU[U_TMASK + (idx >> 5)] = tm;
    atomicAdd(&wsU[U_PCNT + b], (unsigned)__popc(pm));
    atomicAdd(&wsU[U_TCNT + b], (unsigned)__popc(tm));
  }
}

// ------------------- kernel 2: deterministic boundary point extraction -------------------
// Slot s owns flat-index segment [s*SEG, (s+1)*SEG) and takes its first boundary pixel.
__global__ void k_extract(const unsigned* wsU, float* wsF, unsigned* wsUo) {
  const int b     = blockIdx.x >> 1;
  const int which = blockIdx.x & 1;  // 0 = pred map, 1 = target map
  const int maskB = (which ? U_TMASK : U_PMASK) + (b << 15);  // HW/32 = 32768 words/batch
  const unsigned cnt = wsU[(which ? U_TCNT : U_PCNT) + b];
  const int ptsB = (which ? F_TPTS : F_PPTS) + b * SPAD * 2;
  const int valB = (which ? U_TVAL : U_PVAL) + b * SPAD;

  for (int s = threadIdx.x; s < SPAD; s += blockDim.x) {
    float py = 0.0f, px = 0.0f;
    unsigned val = 0u;
    if (s < SPTS && cnt != 0u) {
      const int start = s * SEG;
      if (start < HW) {
        const int end = min(start + SEG, HW);
        int found = -1;
        const int w0 = start >> 5, w1 = (end - 1) >> 5;
        for (int wi = w0; wi <= w1; ++wi) {
          unsigned word = wsU[maskB + wi];
          if (wi == w0) word &= (0xFFFFFFFFu << (start & 31));
          if (wi == w1) { int e = end & 31; if (e) word &= (0xFFFFFFFFu >> (32 - e)); }
          if (word) { found = (wi << 5) + (__ffs((int)word) - 1); break; }
        }
        if (found >= 0) {
          py = (float)(found >> 10);
          px = (float)(found & (Wn - 1));
          val = 1u;
        }
      }
    }
    if (cnt == 0u && s == 0) { py = (float)(Hn / 2); px = (float)(Wn / 2); val = 1u; }  // fallback
    wsF[ptsB + s * 2 + 0] = py;
    wsF[ptsB + s * 2 + 1] = px;
    wsUo[valB + s] = val;
  }
}

// ------------------- kernel 3: directed Hausdorff via WMMA f32 16x16x4 -------------------
// One block per (batch, direction). d^2(m,n) = |a_m|^2 + |b_n|^2 - 2 * (a_m . b_n);
// dot tiles from V_WMMA_F32_16X16X4_F32 (K=2 used: y,x). We keep running MIN over d^2
// (sqrt is monotone -> hoisted out of the inner loop) and encode B-invalidity as
// bn2 = BIGF so no per-iteration validity select is needed.
__global__ void k_hausdorff(const float* wsF, const unsigned* wsU, float* outF) {
  const int b = blockIdx.x >> 1;
  const int d = blockIdx.x & 1;
  const float*    Ap = wsF + (d ? F_TPTS : F_PPTS) + b * SPAD * 2;
  const float*    Bp = wsF + (d ? F_PPTS : F_TPTS) + b * SPAD * 2;
  const unsigned* Av = wsU + (d ? U_TVAL : U_PVAL) + b * SPAD;
  const unsigned* Bv = wsU + (d ? U_PVAL : U_TVAL) + b * SPAD;

  __shared__ float2 sB[SPAD];      // (y,x) per B point
  __shared__ float  sBn2[SPAD];    // |b|^2, or BIGF if invalid
  __shared__ float  minarr[SPAD];
  __shared__ float  red[256];

  const int tid  = threadIdx.x;
  const int wave = tid >> 5;
  const int lane = tid & 31;
  const int l16  = lane & 15;
  const int hi   = lane >> 4;      // 0: lanes 0-15, 1: lanes 16-31

  for (int i = tid; i < SPAD; i += 256) {
    const float by = Bp[i * 2 + 0], bx = Bp[i * 2 + 1];
    sB[i]   = make_float2(by, bx);
    sBn2[i] = Bv[i] ? (by * by + bx * bx) : BIGF;   // invalid -> distance^2 ~ BIGF
    minarr[i] = BIGF;
  }
  __syncthreads();

  for (int mt = wave; mt < NTILE; mt += 8) {
    const int m0 = mt * 16;
    // A fragment: 16x4 f32. lanes 0-15 carry K=0,1 (y,x); lanes 16-31 carry K=2,3 (zero pad)
    v2f a;
    {
      const int m = m0 + l16;
      a[0] = hi ? 0.0f : Ap[m * 2 + 0];
      a[1] = hi ? 0.0f : Ap[m * 2 + 1];
    }
    float an2[8];
    #pragma unroll
    for (int r = 0; r < 8; ++r) {
      const int m = m0 + r + hi * 8;      // C/D layout: VGPR r -> M=r (lanes 0-15) / r+8 (16-31)
      const float ay = Ap[m * 2 + 0], ax = Ap[m * 2 + 1];
      an2[r] = ay * ay + ax * ax;
    }
    float vmin2[8];
    #pragma unroll
    for (int r = 0; r < 8; ++r) vmin2[r] = BIGF;

    for (int nt = 0; nt < NTILE; ++nt) {
      const int n = nt * 16 + l16;
      const float2 bp = sB[n];
      const float  bn2 = sBn2[n];
      v2f bf;
      bf[0] = hi ? 0.0f : bp.x;           // B 4x16: K=0,1 in lanes 0-15; K=2,3 zero
      bf[1] = hi ? 0.0f : bp.y;

      v8f c = {};
      c = __builtin_amdgcn_wmma_f32_16x16x4_f32(false, a, false, bf, (short)0, c, false, false);

      #pragma unroll
      for (int r = 0; r < 8; ++r) {
        const float d2 = (an2[r] + bn2) - 2.0f * c[r];   // huge when B point invalid
        vmin2[r] = fminf(vmin2[r], d2);
      }
    }
    // min across the 16 lanes sharing each output row (two halves independently)
    #pragma unroll
    for (int r = 0; r < 8; ++r) {
      float v = vmin2[r];
      #pragma unroll
      for (int o = 8; o; o >>= 1) v = fminf(v, __shfl_xor(v, o, 16));
      if (l16 == 0) minarr[m0 + r + hi * 8] = sqrtf(fmaxf(v, 0.0f));  // sqrt hoisted here
    }
  }
  __syncthreads();

  // max over valid A points of min-distance
  float vmax = -BIGF;
  for (int i = tid; i < SPAD; i += 256)
    if (Av[i]) vmax = fmaxf(vmax, minarr[i]);
  red[tid] = vmax;
  __syncthreads();
  #pragma unroll
  for (int o = 128; o; o >>= 1) {
    if (tid < o) red[tid] = fmaxf(red[tid], red[tid + o]);
    __syncthreads();
  }
  if (tid == 0) outF[F_AMAX + blockIdx.x] = red[0];
}

// ------------------- kernel 4: final scalar -------------------
__global__ void k_finalize(const float* wsF, float* out) {
  const int b = threadIdx.x;   // 32 threads = one wave
  float hd = fmaxf(wsF[F_AMAX + 2 * b], wsF[F_AMAX + 2 * b + 1]) / DIAG;
  hd = fminf(fmaxf(hd, 0.0f), MAXD);
  const float inter = wsF[F_INTER + b], ps = wsF[F_PSUM + b], ts = wsF[F_TSUM + b];
  const float dice  = (2.0f * inter + SMOOTH) / (ps + ts + SMOOTH);
  float hsum = waveRedAdd(hd);
  float dsum = waveRedAdd(dice);
  if (b == 0) {
    const float invN  = 1.0f / (float)NPIX;
    const float diceL = 1.0f - dsum / (float)Bn;
    const float bndL  = wsF[F_BSUM + 0] * invN + 0.1f * (wsF[F_BSUM + 1] * invN);
    const float hdL   = hsum / (float)Bn;
    out[0] = 0.5f * diceL + 0.3f * bndL + 0.2f * hdL;
  }
}

// ------------------- launch -------------------
extern "C" void kernel_launch(void* const* d_in, const int* in_sizes, int n_in,
                              void* d_out, int out_size, void* d_ws, size_t ws_size,
                              hipStream_t stream) {
  (void)in_sizes; (void)n_in; (void)out_size; (void)ws_size;
  const float* pred = (const float*)d_in[0];
  const float* targ = (const float*)d_in[1];
  float*    wsF = (float*)d_ws;
  unsigned* wsU = (unsigned*)d_ws;
  float*    out = (float*)d_out;

  k_init     <<<1,          512, 0, stream>>>(wsU);
  k_fused    <<<NPIX / 256, 256, 0, stream>>>(pred, targ, wsF, wsU);
  k_extract  <<<Bn * 2,     256, 0, stream>>>(wsU, wsF, wsU);
  k_hausdorff<<<Bn * 2,     256, 0, stream>>>(wsF, wsU, wsF);
  k_finalize <<<1,           32, 0, stream>>>(wsF, out);
}